// GINConvNet_36429912604717
// MI455X (gfx1250) — compile-verified
//
#include <hip/hip_runtime.h>
#include <hip/hip_bf16.h>

// ---------------------------------------------------------------------------
// GIN forward on MI455X (gfx1250): edge-parallel f32 atomics for aggregation,
// V_WMMA_F32_16X16X4_F32 for all GEMMs. Weights are pre-swizzled into WMMA
// B-fragment layout (zero-padded K) so the hot loops are pure
// load_b64 + v_wmma with no divergent guards.
// ---------------------------------------------------------------------------

typedef __attribute__((ext_vector_type(2))) float v2f;
typedef __attribute__((ext_vector_type(8))) float v8f;

#define N_NODES  100000
#define N_EDGES  3200000
#define N_GRAPHS 2048
#define DIM      32
#define FEAT     78
#define FEAT_PAD 80
#define OUT_DIM  128
#define BN_EPS   1e-5f

static __device__ __forceinline__ v8f wmma_f32_16x16x4(v2f a, v2f b, v8f c) {
    // D = A(16x4,f32) * B(4x16,f32) + C(16x16,f32); wave32, EXEC all-1s.
    return __builtin_amdgcn_wmma_f32_16x16x4_f32(
        /*neg_a=*/false, a, /*neg_b=*/false, b,
        /*c_mod=*/(short)0, c, /*reuse_a=*/false, /*reuse_b=*/false);
}

// ---------------------------------------------------------------------------
// Pre-swizzle a row-major weight matrix W[Kreal x N] into WMMA B-fragment
// layout, zero-padded to Kpad rows. One thread per (kf, nh, lane) fragment
// slot; each slot holds float2 {W[kf*4+2*hi][n], W[kf*4+2*hi+1][n]}.
// Packed layout: P[((kf*(N/16) + nh)*32 + lane)*2 + j]
// ---------------------------------------------------------------------------
__global__ void k_repack_w(const float* __restrict__ W, float* __restrict__ P,
                           int Kreal, int Kpad, int N) {
    int t = blockIdx.x * blockDim.x + threadIdx.x;
    int total = (Kpad / 4) * (N / 16) * 32;
    if (t >= total) return;
    int lane = t & 31;
    int rest = t >> 5;
    int nhalves = N / 16;
    int nh = rest % nhalves;
    int kf = rest / nhalves;
    int row0 = kf * 4 + 2 * (lane >> 4);
    int n    = nh * 16 + (lane & 15);
    P[t * 2 + 0] = (row0     < Kreal) ? W[row0 * N + n]       : 0.0f;
    P[t * 2 + 1] = (row0 + 1 < Kreal) ? W[(row0 + 1) * N + n] : 0.0f;
}

// ---------------------------------------------------------------------------
// agg[i][f] = h[i][f] for f < dim, 0 for dim <= f < stride  (zero K-padding)
// ---------------------------------------------------------------------------
__global__ void k_init_agg(const float* __restrict__ h, float* __restrict__ agg,
                           int dim, int stride, int total) {
    int i = blockIdx.x * blockDim.x + threadIdx.x;
    if (i >= total) return;
    int node = i / stride;
    int f    = i - node * stride;
    agg[i] = (f < dim) ? h[node * dim + f] : 0.0f;
}

// ---------------------------------------------------------------------------
// agg[dst[e]] += h[src[e]]   (one wave per edge, lane = feature)
// edge_index is int64 per the reference.
// ---------------------------------------------------------------------------
__global__ __launch_bounds__(256)
void k_scatter(const float* __restrict__ h, const long long* __restrict__ ei,
               float* __restrict__ agg, int dim, int stride, int n_edges) {
    int e    = (int)((blockIdx.x * blockDim.x + threadIdx.x) >> 5);
    int lane = threadIdx.x & 31;
    if (e >= n_edges) return;
    long long s = ei[e];
    long long d = ei[(long long)n_edges + e];
    const float* hs = h + s * dim;
    float*       ad = agg + d * stride;
    for (int f = lane; f < dim; f += 32)
        atomicAdd(&ad[f], hs[f]);
}

// ---------------------------------------------------------------------------
// One wave per 16-node tile:
//   T = relu(AGG[16,Kpad] @ Wa + ba)   (WMMA, pre-swizzled weights)
//   H = relu(T @ Wb + bb)              (WMMA via LDS re-striping)
// writes H (pre-BN) and atomically accumulates per-feature sum / sumsq.
// pkA: Kfrags fragments (each 128 floats: 2 N-halves x 32 lanes x float2)
// pkB: 8 fragments (K=32)
// ---------------------------------------------------------------------------
__global__ __launch_bounds__(256)
void k_gin_mlp(const float* __restrict__ agg, int strideA, int Kfrags,
               const float* __restrict__ pkA, const float* __restrict__ ba,
               const float* __restrict__ pkB, const float* __restrict__ bb,
               float* __restrict__ hout, float* __restrict__ stats, int n_nodes) {
    __shared__ float T[8][16 * DIM];   // per-wave 16x32 staging tile (8 waves)

    const int wid  = threadIdx.x >> 5;
    const int lane = threadIdx.x & 31;
    const int tile = blockIdx.x * 8 + wid;
    if (tile * 16 >= n_nodes) return;          // wave-uniform guard, EXEC stays full
    const int r  = lane & 15;                  // row (A) / column (B,C,D)
    const int hi = lane >> 4;
    const int node0 = tile * 16;
    const int rowbase = (node0 + r) * strideA + 2 * hi;
    const int bslot   = lane * 2;

    // ---------------- GEMM 1: AGG @ Wa ----------------
    v8f acc0 = {}, acc1 = {};
    for (int kf = 0; kf < Kfrags; ++kf) {
        v2f a  = *(const v2f*)&agg[rowbase + kf * 4];
        v2f b0 = *(const v2f*)&pkA[kf * 128 + bslot];
        v2f b1 = *(const v2f*)&pkA[kf * 128 + 64 + bslot];
        acc0 = wmma_f32_16x16x4(a, b0, acc0);
        acc1 = wmma_f32_16x16x4(a, b1, acc1);
    }

    // bias + ReLU, stage to LDS row-major [m][c] for A-fragment reload
    float* Tw = T[wid];
    const float ba0 = ba[r], ba1 = ba[r + 16];
#pragma unroll
    for (int v = 0; v < 8; v++) {
        const int m = v + 8 * hi;
        Tw[m * DIM + r]      = fmaxf(acc0[v] + ba0, 0.0f);
        Tw[m * DIM + r + 16] = fmaxf(acc1[v] + ba1, 0.0f);
    }
    asm volatile("s_wait_dscnt 0" ::: "memory");   // wave-local LDS RAW fence

    // ---------------- GEMM 2: T @ Wb  (K = 32) ----------------
    acc0 = (v8f){}; acc1 = (v8f){};
#pragma unroll
    for (int kf = 0; kf < 8; ++kf) {
        v2f a  = *(const v2f*)&Tw[r * DIM + kf * 4 + 2 * hi];
        v2f b0 = *(const v2f*)&pkB[kf * 128 + bslot];
        v2f b1 = *(const v2f*)&pkB[kf * 128 + 64 + bslot];
        acc0 = wmma_f32_16x16x4(a, b0, acc0);
        acc1 = wmma_f32_16x16x4(a, b1, acc1);
    }

    // bias + ReLU, write H, reduce BN partial sums in-register
    const float bb0 = bb[r], bb1 = bb[r + 16];
    float s0 = 0.f, ss0 = 0.f, s1 = 0.f, ss1 = 0.f;
#pragma unroll
    for (int v = 0; v < 8; v++) {
        const int m = v + 8 * hi;
        const float h0 = fmaxf(acc0[v] + bb0, 0.0f);
        const float h1 = fmaxf(acc1[v] + bb1, 0.0f);
        hout[(node0 + m) * DIM + r]      = h0;
        hout[(node0 + m) * DIM + r + 16] = h1;
        s0 += h0; ss0 += h0 * h0;
        s1 += h1; ss1 += h1 * h1;
    }
    atomicAdd(&stats[r],            s0);
    atomicAdd(&stats[DIM + r],      ss0);
    atomicAdd(&stats[r + 16],       s1);
    atomicAdd(&stats[DIM + r + 16], ss1);
}

// ---------------------------------------------------------------------------
// Fold BN (training mode, biased var) into y = h*A + B per feature.
// ---------------------------------------------------------------------------
__global__ void k_bn_finalize(const float* __restrict__ stats,
                              const float* __restrict__ g,
                              const float* __restrict__ b,
                              float* __restrict__ AB, int n_nodes) {
    int f = threadIdx.x;           // 32 threads
    float inv_n = 1.0f / (float)n_nodes;
    float mu  = stats[f] * inv_n;
    float var = stats[DIM + f] * inv_n - mu * mu;
    float A   = rsqrtf(var + BN_EPS) * g[f];
    AB[f]       = A;
    AB[DIM + f] = b[f] - mu * A;
}

__global__ void k_bn_apply(float* __restrict__ h, const float* __restrict__ AB, int total) {
    int i = blockIdx.x * blockDim.x + threadIdx.x;
    if (i >= total) return;
    int f = i & (DIM - 1);
    h[i] = h[i] * AB[f] + AB[DIM + f];
}

__global__ void k_zero(float* __restrict__ p, int n) {
    int i = blockIdx.x * blockDim.x + threadIdx.x;
    if (i < n) p[i] = 0.0f;
}

// ---------------------------------------------------------------------------
// pooled[batch[i]] += h[i]   (one wave per node, lane = feature)
// ---------------------------------------------------------------------------
__global__ __launch_bounds__(256)
void k_pool(const float* __restrict__ h, const long long* __restrict__ batch,
            float* __restrict__ pooled, int n_nodes) {
    int node = (int)((blockIdx.x * blockDim.x + threadIdx.x) >> 5);
    int lane = threadIdx.x & 31;
    if (node >= n_nodes) return;
    long long g = batch[node];
    atomicAdd(&pooled[g * DIM + lane], h[node * DIM + lane]);
}

// ---------------------------------------------------------------------------
// out[2048,128] = relu(pooled[2048,32] @ Wfc[32,128] + bfc)  (WMMA)
// pkF: fragment (kf, nt) at ((kf*8 + nt)*32 + lane)*2
// ---------------------------------------------------------------------------
__global__ __launch_bounds__(256)
void k_fc(const float* __restrict__ pooled, const float* __restrict__ pkF,
          const float* __restrict__ bfc, float* __restrict__ out, int n_graphs) {
    const int wid  = threadIdx.x >> 5;
    const int lane = threadIdx.x & 31;
    const int tile = blockIdx.x * 8 + wid;
    const int ntiles = OUT_DIM / 16;                  // 8
    if (tile >= (n_graphs / 16) * ntiles) return;     // wave-uniform
    const int mt = tile / ntiles, nt = tile - mt * ntiles;
    const int r  = lane & 15;
    const int hi = lane >> 4;
    const int g0 = mt * 16;
    const int n  = nt * 16 + r;

    v8f acc = {};
#pragma unroll
    for (int kf = 0; kf < 8; ++kf) {
        v2f a = *(const v2f*)&pooled[(g0 + r) * DIM + kf * 4 + 2 * hi];
        v2f b = *(const v2f*)&pkF[(kf * 8 + nt) * 64 + lane * 2];
        acc = wmma_f32_16x16x4(a, b, acc);
    }
    const float bias = bfc[n];
#pragma unroll
    for (int v = 0; v < 8; v++) {
        const int m = v + 8 * hi;
        out[(g0 + m) * OUT_DIM + n] = fmaxf(acc[v] + bias, 0.0f);
    }
}

// ---------------------------------------------------------------------------
extern "C" void kernel_launch(void* const* d_in, const int* in_sizes, int n_in,
                              void* d_out, int out_size, void* d_ws, size_t ws_size,
                              hipStream_t stream) {
    const float*     x     = (const float*)d_in[0];
    const long long* ei    = (const long long*)d_in[1];   // int64 in reference
    const long long* batch = (const long long*)d_in[2];
    const float* W1a = (const float*)d_in[3];
    const float* b1a = (const float*)d_in[4];
    const float* W1b = (const float*)d_in[5];
    const float* b1b = (const float*)d_in[6];
    const float* Wa  = (const float*)d_in[7];    // [4,32,32]
    const float* ba  = (const float*)d_in[8];    // [4,32]
    const float* Wb  = (const float*)d_in[9];    // [4,32,32]
    const float* bb  = (const float*)d_in[10];   // [4,32]
    const float* gamma = (const float*)d_in[11]; // [5,32]
    const float* beta  = (const float*)d_in[12]; // [5,32]
    const float* Wfc = (const float*)d_in[13];   // [32,128]
    const float* bfc = (const float*)d_in[14];   // [128]
    float* out = (float*)d_out;

    // workspace layout (floats)
    float* ws     = (float*)d_ws;
    float* agg    = ws;                                 // 100000*80 = 8,000,000
    float* h0     = agg + (size_t)N_NODES * FEAT_PAD;   // 3,200,000
    float* h1     = h0  + (size_t)N_NODES * DIM;        // 3,200,000
    float* stats  = h1  + (size_t)N_NODES * DIM;        // 5 * 64
    float* bnAB   = stats + 5 * 2 * DIM;                // 5 * 64
    float* pooled = bnAB  + 5 * 2 * DIM;                // 2048*32
    float* pk1a   = pooled + N_GRAPHS * DIM;            // 20*128 = 2560
    float* pk1b   = pk1a + 2560;                        // 8*128  = 1024
    float* pkA    = pk1b + 1024;                        // 4*1024
    float* pkB    = pkA  + 4 * 1024;                    // 4*1024
    float* pkF    = pkB  + 4 * 1024;                    // 8*8*64 = 4096

    // zero atomic accumulator regions (stats + bnAB + pooled are contiguous)
    const int zero_n = 5 * 2 * DIM + 5 * 2 * DIM + N_GRAPHS * DIM;
    k_zero<<<(zero_n + 255) / 256, 256, 0, stream>>>(stats, zero_n);

    // pre-swizzle weights into WMMA B-fragment layout
    k_repack_w<<<(1280 + 255) / 256, 256, 0, stream>>>(W1a, pk1a, FEAT, FEAT_PAD, DIM);
    k_repack_w<<<2, 256, 0, stream>>>(W1b, pk1b, DIM, DIM, DIM);
    for (int i = 0; i < 4; i++) {
        k_repack_w<<<2, 256, 0, stream>>>(Wa + (size_t)i * DIM * DIM, pkA + i * 1024, DIM, DIM, DIM);
        k_repack_w<<<2, 256, 0, stream>>>(Wb + (size_t)i * DIM * DIM, pkB + i * 1024, DIM, DIM, DIM);
    }
    k_repack_w<<<8, 256, 0, stream>>>(Wfc, pkF, DIM, DIM, OUT_DIM);

    const int ntiles      = N_NODES / 16;                 // 6250
    const int mlp_blocks  = (ntiles + 7) / 8;
    const int scat_blocks = (N_EDGES * 32 + 255) / 256;

    // -------- layer 1: 78 -> 32 -> 32 (K padded 78 -> 80) --------
    {
        const int total = N_NODES * FEAT_PAD;
        k_init_agg<<<(total + 255) / 256, 256, 0, stream>>>(x, agg, FEAT, FEAT_PAD, total);
        k_scatter<<<scat_blocks, 256, 0, stream>>>(x, ei, agg, FEAT, FEAT_PAD, N_EDGES);
        k_gin_mlp<<<mlp_blocks, 256, 0, stream>>>(agg, FEAT_PAD, FEAT_PAD / 4,
                                                  pk1a, b1a, pk1b, b1b,
                                                  h0, stats, N_NODES);
        k_bn_finalize<<<1, 32, 0, stream>>>(stats, gamma, beta, bnAB, N_NODES);
        k_bn_apply<<<(N_NODES * DIM + 255) / 256, 256, 0, stream>>>(h0, bnAB, N_NODES * DIM);
    }

    // -------- layers 2..5: 32 -> 32 -> 32 --------
    for (int i = 0; i < 4; i++) {
        float* hin  = (i & 1) ? h1 : h0;
        float* hout = (i & 1) ? h0 : h1;
        const int total = N_NODES * DIM;
        k_init_agg<<<(total + 255) / 256, 256, 0, stream>>>(hin, agg, DIM, DIM, total);
        k_scatter<<<scat_blocks, 256, 0, stream>>>(hin, ei, agg, DIM, DIM, N_EDGES);
        k_gin_mlp<<<mlp_blocks, 256, 0, stream>>>(agg, DIM, DIM / 4,
                                                  pkA + i * 1024, ba + i * DIM,
                                                  pkB + i * 1024, bb + i * DIM,
                                                  hout, stats + (i + 1) * 2 * DIM, N_NODES);
        k_bn_finalize<<<1, 32, 0, stream>>>(stats + (i + 1) * 2 * DIM,
                                            gamma + (i + 1) * DIM, beta + (i + 1) * DIM,
                                            bnAB + (i + 1) * 2 * DIM, N_NODES);
        k_bn_apply<<<(total + 255) / 256, 256, 0, stream>>>(hout, bnAB + (i + 1) * 2 * DIM, total);
    }

    // final node features are in h0 (h0->h1->h0->h1->h0)
    k_pool<<<(N_NODES * 32 + 255) / 256, 256, 0, stream>>>(h0, batch, pooled, N_NODES);

    const int fc_tiles = (N_GRAPHS / 16) * (OUT_DIM / 16);   // 1024 waves
    k_fc<<<(fc_tiles + 7) / 8, 256, 0, stream>>>(pooled, pkF, bfc, out, N_GRAPHS);
}